// TriplaneEncoder_28544352649754
// MI455X (gfx1250) — compile-verified
//
#include <hip/hip_runtime.h>
#include <stdint.h>

#define N_PTS   2097152
#define RES     512
#define C_FEAT  32
#define HW      (RES * RES)
#define PLANE_ELEMS ((size_t)C_FEAT * (size_t)HW)   // 8,388,608 floats per plane

#ifndef USE_ASYNC_LDS
#define USE_ASYNC_LDS 1
#endif

// ---------------------------------------------------------------------------
// Pre-pass: transpose each plane [C=32][512][512] -> [512][512][C=32] so that
// the 32 channels of one texel are a single 128B cacheline. Uses the CDNA5
// async global->LDS DMA path (ASYNCcnt) for the staging load.
// Grid: 3 planes * 512 rows * 16 column-chunks; block (32,32).
// ---------------------------------------------------------------------------
__global__ __launch_bounds__(1024)
void transpose_chw_hwc(const float* __restrict__ Cmat, float* __restrict__ ws)
{
    __shared__ float tile[32][33];                 // pad to 33: conflict-free
    const int t  = blockIdx.x;
    const int cx = t & 15;                         // 32-px chunk in row
    const int y  = (t >> 4) & 511;                 // row
    const int p  = t >> 13;                        // plane 0..2
    const int tx = threadIdx.x;                    // read: pixel | write: channel
    const int ty = threadIdx.y;                    // read: channel | write: pixel

    const float* src = Cmat + (size_t)(p * C_FEAT + ty) * (size_t)HW
                            + (size_t)y * RES + cx * 32 + tx;

#if USE_ASYNC_LDS
    {
        // CDNA5 async DMA: global dword -> LDS address held in vdst, tracked by ASYNCcnt.
        uint32_t lds_off = (uint32_t)(uintptr_t)(&tile[tx][ty]);
        asm volatile("global_load_async_to_lds_b32 %0, %1, off"
                     :: "v"(lds_off), "v"(src) : "memory");
        asm volatile("s_wait_asynccnt 0x0" ::: "memory");
    }
#else
    tile[tx][ty] = *src;
#endif
    __syncthreads();

    // write out pixel-major, channel-contiguous (coalesced over tx)
    const float v = tile[ty][tx];
    ws[((size_t)p * HW + (size_t)y * RES + (size_t)(cx * 32 + ty)) * C_FEAT + tx] = v;
}

// ---------------------------------------------------------------------------
// Main gather kernel (transposed [H][W][C] planes): one wave per point,
// lane = channel. Each bilinear corner = one 128B-aligned cacheline load.
// ---------------------------------------------------------------------------
__device__ __forceinline__ float corner_t(const float* __restrict__ base,
                                          int xi, int yi, float w, int lane)
{
    const bool valid = (xi >= 0) & (xi < RES) & (yi >= 0) & (yi < RES);
    const int  xc = min(max(xi, 0), RES - 1);
    const int  yc = min(max(yi, 0), RES - 1);
    const float v = base[(uint32_t)((yc * RES + xc) * C_FEAT + lane)];
    return v * (valid ? w : 0.0f);
}

__global__ __launch_bounds__(256)
void triplane_gather(const float* __restrict__ x, const float* __restrict__ ws,
                     float* __restrict__ out)
{
    const int lane   = threadIdx.x & 31;
    const int wave   = blockIdx.x * (blockDim.x >> 5) + (threadIdx.x >> 5);
    const int nwaves = gridDim.x * (blockDim.x >> 5);

    for (int i = wave; i < N_PTS; i += nwaves) {
        const float c0 = __builtin_nontemporal_load(&x[3 * i + 0]);
        const float c1 = __builtin_nontemporal_load(&x[3 * i + 1]);
        const float c2 = __builtin_nontemporal_load(&x[3 * i + 2]);
        float acc = 0.0f;
        #pragma unroll
        for (int p = 0; p < 3; ++p) {
            // MAT_IDS = (0,1),(0,2),(1,2): gx indexes width, gy height
            const float gx = (p == 2) ? c1 : c0;
            const float gy = (p == 0) ? c1 : c2;
            const float ix = ((gx + 1.0f) * (float)RES - 1.0f) * 0.5f;
            const float iy = ((gy + 1.0f) * (float)RES - 1.0f) * 0.5f;
            const float x0f = floorf(ix), y0f = floorf(iy);
            const int   x0  = (int)x0f,   y0  = (int)y0f;
            const float wx1 = ix - x0f, wx0 = 1.0f - wx1;
            const float wy1 = iy - y0f, wy0 = 1.0f - wy1;
            const float* base = ws + (size_t)p * PLANE_ELEMS;
            acc += corner_t(base, x0,     y0,     wx0 * wy0, lane);
            acc += corner_t(base, x0 + 1, y0,     wx1 * wy0, lane);
            acc += corner_t(base, x0,     y0 + 1, wx0 * wy1, lane);
            acc += corner_t(base, x0 + 1, y0 + 1, wx1 * wy1, lane);
        }
        // NT store: don't let the 256MB output stream evict the L2-resident planes
        __builtin_nontemporal_store(acc, &out[(size_t)i * C_FEAT + lane]);
    }
}

// ---------------------------------------------------------------------------
// Fallback (workspace too small): direct gather from native [C][H][W] layout.
// ---------------------------------------------------------------------------
__global__ __launch_bounds__(256)
void triplane_gather_direct(const float* __restrict__ x, const float* __restrict__ Cmat,
                            float* __restrict__ out)
{
    const int lane   = threadIdx.x & 31;
    const int wave   = blockIdx.x * (blockDim.x >> 5) + (threadIdx.x >> 5);
    const int nwaves = gridDim.x * (blockDim.x >> 5);

    for (int i = wave; i < N_PTS; i += nwaves) {
        const float c0 = x[3 * i + 0];
        const float c1 = x[3 * i + 1];
        const float c2 = x[3 * i + 2];
        float acc = 0.0f;
        #pragma unroll
        for (int p = 0; p < 3; ++p) {
            const float gx = (p == 2) ? c1 : c0;
            const float gy = (p == 0) ? c1 : c2;
            const float ix = ((gx + 1.0f) * (float)RES - 1.0f) * 0.5f;
            const float iy = ((gy + 1.0f) * (float)RES - 1.0f) * 0.5f;
            const float x0f = floorf(ix), y0f = floorf(iy);
            const int   x0  = (int)x0f,   y0  = (int)y0f;
            const float wx1 = ix - x0f, wx0 = 1.0f - wx1;
            const float wy1 = iy - y0f, wy0 = 1.0f - wy1;
            const float* base = Cmat + (size_t)(p * C_FEAT + lane) * (size_t)HW;
            #pragma unroll
            for (int k = 0; k < 4; ++k) {
                const int  xi = x0 + (k & 1), yi = y0 + (k >> 1);
                const float w = ((k & 1) ? wx1 : wx0) * ((k >> 1) ? wy1 : wy0);
                const bool valid = (xi >= 0) & (xi < RES) & (yi >= 0) & (yi < RES);
                const int  xc = min(max(xi, 0), RES - 1);
                const int  yc = min(max(yi, 0), RES - 1);
                acc += base[(uint32_t)(yc * RES + xc)] * (valid ? w : 0.0f);
            }
        }
        __builtin_nontemporal_store(acc, &out[(size_t)i * C_FEAT + lane]);
    }
}

// ---------------------------------------------------------------------------
extern "C" void kernel_launch(void* const* d_in, const int* in_sizes, int n_in,
                              void* d_out, int out_size, void* d_ws, size_t ws_size,
                              hipStream_t stream)
{
    const float* x    = (const float*)d_in[0];   // [N_PTS, 3]
    const float* Cmat = (const float*)d_in[1];   // [3, 32, 512, 512]
    float*       out  = (float*)d_out;           // [N_PTS, 32]

    const size_t need = 3 * PLANE_ELEMS * sizeof(float);   // 96 MB
    if (ws_size >= need) {
        float* ws = (float*)d_ws;
        dim3 tb(32, 32);
        transpose_chw_hwc<<<3 * 512 * 16, tb, 0, stream>>>(Cmat, ws);
        triplane_gather<<<8192, 256, 0, stream>>>(x, ws, out);
    } else {
        triplane_gather_direct<<<8192, 256, 0, stream>>>(x, Cmat, out);
    }
}